// Response_57466662421175
// MI455X (gfx1250) — compile-verified
//
#include <hip/hip_runtime.h>
#include <hip/hip_bf16.h>
#include <math.h>

// ---------------- problem constants (from reference) ----------------
#define BATCH      8
#define LR         128          // decoder length
#define HDIM       512          // hidden
#define CTX        256
#define STATE      32
#define ACT        32
#define LSRC       320          // CTX+STATE+ACT
#define NVOCAB     32000
#define NROWS      (BATCH * LR) // 1024
#define SCALE      0.044194173824159216f  // 1/sqrt(512)

typedef __attribute__((ext_vector_type(2))) float v2f;
typedef __attribute__((ext_vector_type(8))) float v8f;

// ---------------- async-LDS path: inline CDNA5 asm (gfx1250 device pass only) ----
#if defined(__gfx1250__)
#define USE_ASYNC_LDS 1
#else
#define USE_ASYNC_LDS 0
#endif

#if USE_ASYNC_LDS
// dst: LDS byte offset (low 32 bits of the flat shared pointer), src: global addr
__device__ __forceinline__ void async_load_b128(const float* gp, float* lp) {
  const unsigned ldsa = (unsigned)(size_t)lp;   // flat LDS addr low 32 bits = LDS offset
  asm volatile("global_load_async_to_lds_b128 %0, %1, off"
               :: "v"(ldsa), "v"(gp) : "memory");
}
__device__ __forceinline__ void async_wait0() {
#if __has_builtin(__builtin_amdgcn_s_wait_asynccnt)
  __builtin_amdgcn_s_wait_asynccnt(0);
#else
  asm volatile("s_wait_asynccnt 0" ::: "memory");
#endif
}
#endif

// =====================================================================
// Kernel 1: copy_logit[b,l,s] = scale * mask(dot(dec[b,l,:], src[b,s,:]))
// One wave per 16x16 (l,s) tile. WMMA f32 16x16x4, K loop over 512.
// A fragment (16x4): lane<16 -> row l0+lane, cols k,k+1 ; lane>=16 -> k+2,k+3
// B fragment (4x16): lane<16 -> col s0+lane from rows k,k+1 ; lane>=16 -> k+2,k+3
//   (src_hidden row s is contiguous in h, so this is a float2 load per lane)
// =====================================================================
__global__ __launch_bounds__(32) void copy_logit_wmma(
    const float* __restrict__ dec,      // (B, LR, H)
    const float* __restrict__ srch,     // (B, LSRC, H)
    const int*   __restrict__ mask,     // (B, 1, LSRC)
    float*       __restrict__ cws)      // (B, LR, LSRC)
{
  const int lane = threadIdx.x & 31;
  const int half = lane >> 4;     // 0: lanes 0-15, 1: lanes 16-31
  const int lr   = lane & 15;
  const int s0 = blockIdx.x * 16;
  const int l0 = blockIdx.y * 16;
  const int b  = blockIdx.z;

  const float* ap = dec  + ((size_t)b * LR   + l0 + lr) * HDIM + 2 * half;
  const float* bp = srch + ((size_t)b * LSRC + s0 + lr) * HDIM + 2 * half;

  v8f acc = {};
#pragma unroll 8
  for (int k = 0; k < HDIM; k += 4) {
    v2f a; a.x = ap[k]; a.y = ap[k + 1];
    v2f bb; bb.x = bp[k]; bb.y = bp[k + 1];
    acc = __builtin_amdgcn_wmma_f32_16x16x4_f32(false, a, false, bb,
                                                (short)0, acc, false, false);
  }

  const int s   = s0 + lr;
  const int msk = mask[(size_t)b * LSRC + s];
#pragma unroll
  for (int r = 0; r < 8; ++r) {
    float v = acc[r];
    if (msk == 0) v = -1.0e9f;               // mask before scaling (as reference)
    const int row = l0 + r + 8 * half;       // C/D layout: VGPR r = M r (+8 for hi half)
    cws[(((size_t)b * LR) + row) * LSRC + s] = v * SCALE;
  }
}

// =====================================================================
// Kernel 2: gen logits GEMM: out[m, n] = scale * (dec[m,:] @ Wg[:,n] + bg[n])
// Block = 256 threads (8 waves). Block tile M=32 x N=256, K chunk = 32.
// Wave (w): rows 16*(w>>2), cols 64*(w&3) .. +63 (4 accumulators of 16x16).
// W_gen K-chunk staged in LDS via global_load_async_to_lds_b128 (tracked by
// ASYNCcnt, drained with s_wait_asynccnt). Padded row stride of 272 floats
// puts lanes 0-15 and 16-31 on disjoint LDS banks during fragment reads.
// =====================================================================
#define BM 32
#define BN 256
#define BK 32
#define LDSB_STRIDE 272

__global__ __launch_bounds__(256) void gen_logit_gemm(
    const float* __restrict__ dec,   // (NROWS, H)
    const float* __restrict__ Wg,    // (H, NVOCAB)
    const float* __restrict__ bg,    // (NVOCAB)
    float*       __restrict__ out)   // (NROWS, NVOCAB) -> scaled logits
{
  __shared__ __align__(16) float ldsB[BK * LDSB_STRIDE];

  const int tid  = threadIdx.x;
  const int lane = tid & 31;
  const int wave = tid >> 5;
  const int half = lane >> 4;
  const int lr   = lane & 15;

  const int m0 = blockIdx.y * BM + 16 * (wave >> 2);
  const int n0 = blockIdx.x * BN;
  const int wn = 64 * (wave & 3);

  v8f acc[4] = {};

  for (int k0 = 0; k0 < HDIM; k0 += BK) {
    // ---- stage B chunk (BK x BN = 32x256 f32) into LDS, b128 per lane ----
#pragma unroll
    for (int i = 0; i < 8; ++i) {
      const int f   = i * 256 + tid;     // float4 index in chunk (2048 total)
      const int row = f >> 6;            // 64 float4 per 256-wide row
      const int c4  = f & 63;
      const float* gp = Wg + (size_t)(k0 + row) * NVOCAB + n0 + c4 * 4;
      float* lp = &ldsB[row * LDSB_STRIDE + c4 * 4];
#if USE_ASYNC_LDS
      async_load_b128(gp, lp);
#else
      *(float4*)lp = *(const float4*)gp;
#endif
    }
#if USE_ASYNC_LDS
    async_wait0();
#endif
    __syncthreads();

    // ---- WMMA over the K chunk ----
    const float* ap = dec + (size_t)(m0 + lr) * HDIM + k0 + 2 * half;
#pragma unroll
    for (int kk = 0; kk < BK; kk += 4) {
      v2f a; a.x = ap[kk]; a.y = ap[kk + 1];
#pragma unroll
      for (int j = 0; j < 4; ++j) {
        const float* bp = &ldsB[(kk + 2 * half) * LDSB_STRIDE + wn + 16 * j + lr];
        v2f b; b.x = bp[0]; b.y = bp[LDSB_STRIDE];
        acc[j] = __builtin_amdgcn_wmma_f32_16x16x4_f32(false, a, false, b,
                                                       (short)0, acc[j], false, false);
      }
    }
    __syncthreads();
  }

  // ---- epilogue: (+bias) * scale ----
#pragma unroll
  for (int j = 0; j < 4; ++j) {
    const int col = n0 + wn + 16 * j + lr;
    const float bia = bg[col];
#pragma unroll
    for (int r = 0; r < 8; ++r) {
      const int row = m0 + r + 8 * half;
      out[(size_t)row * NVOCAB + col] = (acc[j][r] + bia) * SCALE;
    }
  }
}

// =====================================================================
// Kernel 3: per-row online softmax stats over 32000 gen + 320 copy logits.
// stats[row] = (rowmax, 1/rowsum)
// =====================================================================
__global__ __launch_bounds__(256) void softmax_stats(
    const float* __restrict__ gen,   // (NROWS, NVOCAB) scaled logits
    const float* __restrict__ cws,   // (NROWS, LSRC)   scaled masked logits
    float2*      __restrict__ stats) // (NROWS)
{
  const int row = blockIdx.x;
  const int tid = threadIdx.x;
  const float* g = gen + (size_t)row * NVOCAB;
  const float* c = cws + (size_t)row * LSRC;

  float m = -3.4e38f, s = 0.f;
  for (int i = tid; i < NVOCAB; i += 256) {
    const float x = g[i];
    const float nm = fmaxf(m, x);
    s = s * __expf(m - nm) + __expf(x - nm);
    m = nm;
  }
  for (int i = tid; i < LSRC; i += 256) {
    const float x = c[i];
    const float nm = fmaxf(m, x);
    s = s * __expf(m - nm) + __expf(x - nm);
    m = nm;
  }

  __shared__ float sm[256];
  __shared__ float ss[256];
  sm[tid] = m; ss[tid] = s;
  __syncthreads();
  for (int off = 128; off > 0; off >>= 1) {
    if (tid < off) {
      const float m2 = sm[tid + off], s2 = ss[tid + off];
      const float nm = fmaxf(sm[tid], m2);
      ss[tid] = ss[tid] * __expf(sm[tid] - nm) + s2 * __expf(m2 - nm);
      sm[tid] = nm;
    }
    __syncthreads();
  }
  if (tid == 0) stats[row] = make_float2(sm[0], 1.f / ss[0]);
}

// =====================================================================
// Kernel 4: finalize gen probabilities in place.
// =====================================================================
__global__ __launch_bounds__(256) void finalize_gen(
    float* __restrict__ out, const float2* __restrict__ stats)
{
  const size_t idx = (size_t)blockIdx.x * 256 + threadIdx.x;
  const size_t total = (size_t)NROWS * NVOCAB;
  if (idx >= total) return;
  const int row = (int)(idx / NVOCAB);
  const float2 st = stats[row];
  out[idx] = __expf(out[idx] - st.x) * st.y;
}

// =====================================================================
// Kernel 5: scatter copy probabilities into vocab (atomicAdd per entry).
// =====================================================================
__global__ __launch_bounds__(320) void scatter_copy(
    float*       __restrict__ out,
    const float* __restrict__ cws,
    const float2* __restrict__ stats,
    const int*   __restrict__ context,  // (B, CTX)
    const int*   __restrict__ tp,       // (B, STATE)
    const int*   __restrict__ action,   // (B, ACT)
    const int*   __restrict__ loc2glo)  // (NTOPIC)
{
  const int row = blockIdx.x;
  const int s   = threadIdx.x;           // 0..319
  const int b   = row >> 7;              // row / LR
  const float2 st = stats[row];
  const float p = __expf(cws[(size_t)row * LSRC + s] - st.x) * st.y;

  int col;
  if (s < CTX)              col = context[b * CTX + s];
  else if (s < CTX + STATE) col = loc2glo[tp[b * STATE + (s - CTX)]];
  else                      col = loc2glo[action[b * ACT + (s - CTX - STATE)]];

  atomicAdd(out + (size_t)row * NVOCAB + col, p);
}

// =====================================================================
// launch
// =====================================================================
extern "C" void kernel_launch(void* const* d_in, const int* in_sizes, int n_in,
                              void* d_out, int out_size, void* d_ws, size_t ws_size,
                              hipStream_t stream) {
  const float* dec      = (const float*)d_in[0];   // (B, LR, H)
  const float* srch     = (const float*)d_in[1];   // (B, LSRC, H)
  const float* Wg       = (const float*)d_in[2];   // (H, NVOCAB)
  const float* bg       = (const float*)d_in[3];   // (NVOCAB)
  const int*   src_mask = (const int*)d_in[4];     // (B, 1, LSRC)
  const int*   context  = (const int*)d_in[5];     // (B, CTX)
  const int*   tp       = (const int*)d_in[6];     // (B, STATE)
  const int*   action   = (const int*)d_in[7];     // (B, ACT)
  const int*   loc2glo  = (const int*)d_in[8];     // (NTOPIC)

  float* out = (float*)d_out;                      // (B, LR, NVOCAB)

  // workspace layout
  float*  cws   = (float*)d_ws;                               // NROWS*LSRC f32
  float2* stats = (float2*)((char*)d_ws +
                            (size_t)NROWS * LSRC * sizeof(float)); // NROWS f32x2

  // 1. copy logits (WMMA, per-batch attention scores), masked + scaled
  copy_logit_wmma<<<dim3(LSRC / 16, LR / 16, BATCH), 32, 0, stream>>>(
      dec, srch, src_mask, cws);

  // 2. gen logits GEMM (WMMA), bias + scale, into d_out
  gen_logit_gemm<<<dim3(NVOCAB / BN, NROWS / BM), 256, 0, stream>>>(
      dec, Wg, bg, out);

  // 3. softmax stats over the concatenated 32320-wide row
  softmax_stats<<<NROWS, 256, 0, stream>>>(out, cws, stats);

  // 4. finalize gen probabilities in place
  {
    const size_t total = (size_t)NROWS * NVOCAB;
    const int blocks = (int)((total + 255) / 256);
    finalize_gen<<<blocks, 256, 0, stream>>>(out, stats);
  }

  // 5. scatter copy probabilities
  scatter_copy<<<NROWS, LSRC, 0, stream>>>(out, cws, stats,
                                           context, tp, action, loc2glo);
}